// DotInteraction_15745350107452
// MI455X (gfx1250) — compile-verified
//
#include <hip/hip_runtime.h>

typedef __attribute__((ext_vector_type(2))) float v2f;
typedef __attribute__((ext_vector_type(4))) float v4f;
typedef __attribute__((ext_vector_type(8))) float v8f;

#define NF 64
#define ND 128
#define LDS_STRIDE 132     // 128 + 4 floats pad -> conflict-free column reads
#define NPAIR 2016         // 64*63/2

// D = A(16x4) * B(4x16) + C, f32 WMMA
#define WMMA_F32(A, B, C) \
  __builtin_amdgcn_wmma_f32_16x16x4_f32(false, (A), false, (B), (short)0, (C), false, false)

__device__ __forceinline__ void store_tile(float* __restrict__ outb, const v8f c,
                                           int ti, int tj, int lrow, int khalf) {
  const int j = tj * 16 + lrow;      // N = lane % 16
#pragma unroll
  for (int v = 0; v < 8; ++v) {
    const int i = ti * 16 + v + 8 * khalf;   // M = vgpr + 8*(lane>=16)
    if (i < j) {
      const int idx = i * 63 - (i * (i - 1)) / 2 + (j - i - 1);
      outb[idx] = c[v];
    }
  }
}

__global__ __launch_bounds__(128) void gram_triu_wmma(const float* __restrict__ x,
                                                      float* __restrict__ out) {
  __shared__ float smem[NF * LDS_STRIDE];

  const int b = blockIdx.x;
  const int t = threadIdx.x;                       // 0..127
  const float* __restrict__ xb = x + (size_t)b * (NF * ND);
  float* __restrict__ outb = out + (size_t)b * NPAIR;

  // ---- Stage x[b] (64x128 f32 = 32KB) into LDS, coalesced 16B loads ----
  // float4 index j = i*128 + t : each iteration reads 2KB contiguous.
#pragma unroll
  for (int i = 0; i < 16; ++i) {
    const int j = i * 128 + t;                     // 0..2047
    const v4f v = ((const v4f*)xb)[j];
    const int row = j >> 5;                        // (j*4)/128
    const int col = (j & 31) << 2;
    float* dst = &smem[row * LDS_STRIDE + col];
    dst[0] = v.x; dst[1] = v.y; dst[2] = v.z; dst[3] = v.w;
  }
  __syncthreads();

  const int wave  = t >> 5;                        // wave32: 4 waves/block
  const int lane  = t & 31;
  const int lrow  = lane & 15;                     // M (for A) / N (for B)
  const int khalf = lane >> 4;                     // K offset 0 or 2

  // Fragment of row-band `ti` for K-chunk `kc`: serves as both A and B operand.
#define LOAD_FRAG(ti, kc) \
  (*(const v2f*)&smem[((ti) * 16 + lrow) * LDS_STRIDE + (kc) * 4 + khalf * 2])

  // 10 upper-tri 16x16 tiles split {3,3,2,2} over 4 waves (wave-uniform branch,
  // EXEC stays all-ones around every WMMA).
  v8f a0 = {}, a1 = {}, a2 = {};

  switch (wave) {
    case 0: {  // (0,0) (1,1) (2,3)
#pragma unroll 4
      for (int kc = 0; kc < ND / 4; ++kc) {
        v2f f0 = LOAD_FRAG(0, kc), f1 = LOAD_FRAG(1, kc);
        v2f f2 = LOAD_FRAG(2, kc), f3 = LOAD_FRAG(3, kc);
        a0 = WMMA_F32(f0, f0, a0);
        a1 = WMMA_F32(f1, f1, a1);
        a2 = WMMA_F32(f2, f3, a2);
      }
      store_tile(outb, a0, 0, 0, lrow, khalf);
      store_tile(outb, a1, 1, 1, lrow, khalf);
      store_tile(outb, a2, 2, 3, lrow, khalf);
      break;
    }
    case 1: {  // (0,1) (1,2) (3,3)
#pragma unroll 4
      for (int kc = 0; kc < ND / 4; ++kc) {
        v2f f0 = LOAD_FRAG(0, kc), f1 = LOAD_FRAG(1, kc);
        v2f f2 = LOAD_FRAG(2, kc), f3 = LOAD_FRAG(3, kc);
        a0 = WMMA_F32(f0, f1, a0);
        a1 = WMMA_F32(f1, f2, a1);
        a2 = WMMA_F32(f3, f3, a2);
      }
      store_tile(outb, a0, 0, 1, lrow, khalf);
      store_tile(outb, a1, 1, 2, lrow, khalf);
      store_tile(outb, a2, 3, 3, lrow, khalf);
      break;
    }
    case 2: {  // (0,2) (1,3)
#pragma unroll 4
      for (int kc = 0; kc < ND / 4; ++kc) {
        v2f f0 = LOAD_FRAG(0, kc), f1 = LOAD_FRAG(1, kc);
        v2f f2 = LOAD_FRAG(2, kc), f3 = LOAD_FRAG(3, kc);
        a0 = WMMA_F32(f0, f2, a0);
        a1 = WMMA_F32(f1, f3, a1);
      }
      store_tile(outb, a0, 0, 2, lrow, khalf);
      store_tile(outb, a1, 1, 3, lrow, khalf);
      break;
    }
    default: {  // (0,3) (2,2)
#pragma unroll 4
      for (int kc = 0; kc < ND / 4; ++kc) {
        v2f f0 = LOAD_FRAG(0, kc), f2 = LOAD_FRAG(2, kc);
        v2f f3 = LOAD_FRAG(3, kc);
        a0 = WMMA_F32(f0, f3, a0);
        a1 = WMMA_F32(f2, f2, a1);
      }
      store_tile(outb, a0, 0, 3, lrow, khalf);
      store_tile(outb, a1, 2, 2, lrow, khalf);
      break;
    }
  }
#undef LOAD_FRAG
}

extern "C" void kernel_launch(void* const* d_in, const int* in_sizes, int n_in,
                              void* d_out, int out_size, void* d_ws, size_t ws_size,
                              hipStream_t stream) {
  const float* x = (const float*)d_in[0];
  float* out = (float*)d_out;
  const int B = in_sizes[0] / (NF * ND);           // 16384
  gram_triu_wmma<<<dim3(B), dim3(128), 0, stream>>>(x, out);
}